// MPNN_25761213841966
// MI455X (gfx1250) — compile-verified
//
#include <hip/hip_runtime.h>
#include <hip/hip_bf16.h>

// Memory-bound MPNN message-pass:
//   msg = vec * pv1[0] + pv2[0] * edge_vec   (elementwise, [E,3,128])
//   agg = scatter_add(msg, tgt)              (fp32 atomics, L2-resident)
//   out = LayerNorm_128(agg[src])            (wave32: 1 wave per row)
//   out duplicated to both n_v slots.

typedef float v4f __attribute__((ext_vector_type(4)));
typedef int   v4i __attribute__((ext_vector_type(4)));

#define AS1 __attribute__((address_space(1)))
#define AS3 __attribute__((address_space(3)))

#ifndef __has_builtin
#define __has_builtin(x) 0
#endif

#if __has_builtin(__builtin_amdgcn_global_load_async_to_lds_b128) && \
    __has_builtin(__builtin_amdgcn_s_wait_asynccnt)
#define USE_ASYNC_LDS 1
#else
#define USE_ASYNC_LDS 0
#endif

// ---------------------------------------------------------------------------
// Kernel 1: scatter phase. One thread per float4 of msg (E*96 threads).
// vec / pos_vec streamed once -> NT loads (don't thrash L2); the accumulator
// (~15 MB hot rows) takes default RT policy so atomics resolve in L2.
// ---------------------------------------------------------------------------
__global__ __launch_bounds__(256) void mpnn_scatter(
    const float* __restrict__ vec, const float* __restrict__ pv1,
    const float* __restrict__ pv2, const float* __restrict__ ev,
    const int* __restrict__ eidx, float* __restrict__ agg, int E)
{
    unsigned t = blockIdx.x * 256u + threadIdx.x;
    unsigned total = (unsigned)E * 96u;        // 96 float4 per edge (3*128/4)
    if (t >= total) return;
    unsigned e  = t / 96u;
    unsigned q  = t - e * 96u;                 // 0..95: float4 index in [3,128]
    unsigned c  = q >> 5;                      // component 0..2
    unsigned d4 = q & 31u;                     // float4 index within D

    v4f v  = __builtin_nontemporal_load((const v4f*)(vec + (size_t)e * 384u + q  * 4u));
    v4f p1 = __builtin_nontemporal_load((const v4f*)(pv1 + (size_t)e * 128u + d4 * 4u));
    v4f p2 = __builtin_nontemporal_load((const v4f*)(pv2 + (size_t)e * 128u + d4 * 4u));
    float evc = ev[e * 3u + c];                // broadcast within 32 lanes

    v4f m = v * p1 + p2 * evc;

    int tgt = eidx[2u * e + 1u];
    float* dst = agg + (size_t)tgt * 384u + q * 4u;
    atomicAdd(dst + 0, m.x);
    atomicAdd(dst + 1, m.y);
    atomicAdd(dst + 2, m.z);
    atomicAdd(dst + 3, m.w);
}

// ---------------------------------------------------------------------------
// Kernel 2: gather + LayerNorm + duplicated store. One wave32 per (e,c) row:
// 32 lanes x float4 == 128 elements == one LN row == one 512B cacheline run.
// agg row is staged through LDS with the CDNA5 async-copy path when available;
// output is written twice with NT stores (streamed, never re-read).
// ---------------------------------------------------------------------------
__global__ __launch_bounds__(256) void mpnn_gather_ln(
    const int* __restrict__ eidx, const float* __restrict__ agg,
    const float* __restrict__ gamma, const float* __restrict__ beta,
    float* __restrict__ out, int E)
{
#if USE_ASYNC_LDS
    __shared__ v4f stage[8 * 32];              // 8 waves x 512B
#endif
    unsigned lane = threadIdx.x & 31u;
    unsigned wv   = threadIdx.x >> 5;
    unsigned row  = blockIdx.x * 8u + wv;      // row in [0, E*3)
    unsigned nrows = (unsigned)E * 3u;
    if (row >= nrows) return;                  // wave-uniform exit: EXEC all-1s below

    unsigned e = row / 3u;
    unsigned c = row - e * 3u;
    int src = eidx[2u * e + 0u];
    const float* grow = agg + (size_t)src * 384u + c * 128u + lane * 4u;

    v4f x;
#if USE_ASYNC_LDS
    {
        AS1 v4i* gsrc = (AS1 v4i*)((AS1 void*)grow);
        AS3 v4i* ldst = (AS3 v4i*)((AS3 void*)&stage[wv * 32u + lane]);
        __builtin_amdgcn_global_load_async_to_lds_b128(gsrc, ldst, 0, 0);
        __builtin_amdgcn_s_wait_asynccnt(0);
        x = stage[wv * 32u + lane];
    }
#else
    x = *(const v4f*)grow;
#endif

    // mean over 128: local 4-sum + 5-step wave32 butterfly
    float s = x.x + x.y + x.z + x.w;
#pragma unroll
    for (int off = 16; off >= 1; off >>= 1) s += __shfl_xor(s, off, 32);
    float mu = s * (1.0f / 128.0f);

    float dx = x.x - mu, dy = x.y - mu, dz = x.z - mu, dw = x.w - mu;
    float sq = dx * dx + dy * dy + dz * dz + dw * dw;
#pragma unroll
    for (int off = 16; off >= 1; off >>= 1) sq += __shfl_xor(sq, off, 32);
    float rs = rsqrtf(sq * (1.0f / 128.0f) + 1e-5f);

    v4f g = *(const v4f*)(gamma + lane * 4u);
    v4f b = *(const v4f*)(beta  + lane * 4u);
    v4f y;
    y.x = dx * rs * g.x + b.x;
    y.y = dy * rs * g.y + b.y;
    y.z = dz * rs * g.z + b.z;
    y.w = dw * rs * g.w + b.w;

    size_t o = (size_t)row * 128u + lane * 4u;
    __builtin_nontemporal_store(y, (v4f*)(out + o));                         // n_v slot 0
    __builtin_nontemporal_store(y, (v4f*)(out + (size_t)E * 384u + o));      // n_v slot 1
}

// ---------------------------------------------------------------------------
extern "C" void kernel_launch(void* const* d_in, const int* in_sizes, int n_in,
                              void* d_out, int out_size, void* d_ws, size_t ws_size,
                              hipStream_t stream) {
    const int*   eidx  = (const int*)d_in[0];    // [E,2]
    const float* vec   = (const float*)d_in[1];  // [E,3,128]
    const float* pv1   = (const float*)d_in[2];  // [2,E,128] -> use slice 0
    const float* pv2   = (const float*)d_in[3];  // [2,E,128] -> use slice 0
    const float* ev    = (const float*)d_in[4];  // [E,3]
    const float* gamma = (const float*)d_in[5];  // [128]
    const float* beta  = (const float*)d_in[6];  // [128]

    const int E = in_sizes[0] / 2;
    float* agg = (float*)d_ws;                   // [E, 384] accumulator
    const size_t aggBytes = (size_t)E * 384u * sizeof(float);

    (void)hipMemsetAsync(agg, 0, aggBytes, stream);   // capture-safe zeroing

    {
        unsigned total = (unsigned)E * 96u;
        mpnn_scatter<<<dim3((total + 255u) / 256u), dim3(256), 0, stream>>>(
            vec, pv1, pv2, ev, eidx, agg, E);
    }
    {
        unsigned rows = (unsigned)E * 3u;
        mpnn_gather_ln<<<dim3((rows + 7u) / 8u), dim3(256), 0, stream>>>(
            eidx, agg, gamma, beta, (float*)d_out, E);
    }
}